// SimpleGCN_27788438405232
// MI455X (gfx1250) — compile-verified
//
#include <hip/hip_runtime.h>
#include <math.h>

// Problem constants (match reference)
#define N_NODES  100000
#define N_EDGES  3200000
#define F_IN     128
#define HIDDEN   32
#define N_GRAPHS 256

typedef __attribute__((ext_vector_type(2))) float v2f;
typedef __attribute__((ext_vector_type(8))) float v8f;

// ---------------- utility kernels ----------------

__global__ void fill_kernel(float* __restrict__ p, float v, int n) {
    int i = blockIdx.x * blockDim.x + threadIdx.x;
    if (i < n) p[i] = v;
}

__global__ void deg_accum_kernel(const int* __restrict__ dst, float* __restrict__ deg, int E) {
    int e = blockIdx.x * blockDim.x + threadIdx.x;
    if (e < E) atomicAdd(&deg[dst[e]], 1.0f);
}

__global__ void rsqrt_kernel(float* __restrict__ deg, int n) {
    int i = blockIdx.x * blockDim.x + threadIdx.x;
    if (i < n) {
        float d = deg[i];
        deg[i] = (d > 0.0f) ? rsqrtf(d) : 0.0f;
    }
}

// ---------------- fp32 WMMA GEMM: C[M x 32] = act(A[M x K] + biasA) * B[K x 32] ----------
// One wave computes a 16x32 panel using V_WMMA_F32_16X16X4_F32 (exact fp32).
// A 16x4 frag: lanes 0-15 hold (K=k0,k0+1), lanes 16-31 hold (K=k0+2,k0+3), row = lane&15.
// B 4x16 frag: same K split, col = lane&15.
// C 16x16: VGPR r holds rows r (lanes 0-15) and r+8 (lanes 16-31), col = lane&15.
// BIAS_RELU_A fuses the conv1 epilogue relu(a + b1[k]) into the A-operand load
// (each A element is consumed exactly once, so the fusion is exact and saves a full pass).
template <int K, bool BIAS_RELU_A>
__global__ void gemm_n32_wmma_f32(const float* __restrict__ A,
                                  const float* __restrict__ B,
                                  float* __restrict__ C, int M,
                                  const float* __restrict__ biasA) {
    const int lane  = threadIdx.x & 31;
    const int wave  = threadIdx.x >> 5;
    const int tile  = blockIdx.x * (blockDim.x >> 5) + wave;
    const int ntile = M >> 4;               // M is a multiple of 16 (100000 = 6250*16)
    if (tile >= ntile) return;              // wave-uniform exit (EXEC all-1s for WMMA)

    const int m0    = tile << 4;
    const int rlane = lane & 15;
    const int khalf = (lane >> 4) << 1;     // 0 for lanes 0-15, 2 for lanes 16-31

    v8f c0 = {};                            // N-tile 0 (cols 0..15)
    v8f c1 = {};                            // N-tile 1 (cols 16..31)

    const float* arow = A + (size_t)(m0 + rlane) * K;

#pragma unroll
    for (int k0 = 0; k0 < K; k0 += 4) {
        float a0 = arow[k0 + khalf];
        float a1 = arow[k0 + khalf + 1];
        if (BIAS_RELU_A) {
            a0 = fmaxf(a0 + biasA[k0 + khalf], 0.0f);
            a1 = fmaxf(a1 + biasA[k0 + khalf + 1], 0.0f);
        }
        v2f a; a.x = a0; a.y = a1;

        const float* bp = B + (size_t)(k0 + khalf) * 32 + rlane;
        v2f b0, b1;
        b0.x = bp[0];   b0.y = bp[32];      // cols 0..15,  K and K+1
        b1.x = bp[16];  b1.y = bp[48];      // cols 16..31, K and K+1

        c0 = __builtin_amdgcn_wmma_f32_16x16x4_f32(false, a, false, b0, (short)0, c0, false, false);
        c1 = __builtin_amdgcn_wmma_f32_16x16x4_f32(false, a, false, b1, (short)0, c1, false, false);
    }

    const int mbase = (lane >> 4) << 3;     // 0 or 8
#pragma unroll
    for (int r = 0; r < 8; ++r) {
        float* crow = C + (size_t)(m0 + mbase + r) * 32;
        crow[rlane]      = c0[r];
        crow[rlane + 16] = c1[r];
    }
}

// ---------------- normalized gather + scatter-add over edges + self-loops ----------------
// One wave per edge, one lane per feature. The edge id and node indices are
// wave-uniform: force them into SGPRs so index/dinv traffic becomes scalar
// loads (4 s_loads per wave instead of 4 vector loads per lane).
__global__ void gcn_scatter_kernel(const int* __restrict__ src, const int* __restrict__ dst,
                                   const float* __restrict__ dinv,
                                   const float* __restrict__ feat, float* __restrict__ out,
                                   int E, int N) {
    int tid  = blockIdx.x * blockDim.x + threadIdx.x;   // (E+N)*32 = 105.6M fits in int
    int e    = tid >> 5;
    int lane = threadIdx.x & 31;
    if (e >= E + N) return;
    e = __builtin_amdgcn_readfirstlane(e);
    int s, d;
    if (e < E) {
        s = __builtin_amdgcn_readfirstlane(src[e]);
        d = __builtin_amdgcn_readfirstlane(dst[e]);
    } else {
        s = d = e - E;                                  // self-loop (already scalar)
    }
    float w = dinv[s] * dinv[d];                        // scalar loads, broadcast multiply
    atomicAdd(&out[((size_t)d << 5) + lane], feat[((size_t)s << 5) + lane] * w);
}

// ---------------- pooling ----------------
__global__ void counts_kernel(const int* __restrict__ batch, float* __restrict__ counts, int n) {
    int i = blockIdx.x * blockDim.x + threadIdx.x;
    if (i < n) atomicAdd(&counts[batch[i]], 1.0f);
}

__global__ void pool_sum_kernel(const int* __restrict__ batch, const float* __restrict__ feat,
                                float* __restrict__ pooled, int n) {
    int tid  = blockIdx.x * blockDim.x + threadIdx.x;
    int node = tid >> 5;
    int f    = threadIdx.x & 31;
    if (node >= n) return;
    node = __builtin_amdgcn_readfirstlane(node);
    int g = __builtin_amdgcn_readfirstlane(batch[node]);   // wave-uniform graph id
    atomicAdd(&pooled[(g << 5) + f], feat[((size_t)node << 5) + f]);
}

// ---------------- MLP head: mean, +b2, relu(p@Wm1+bm1)@Wm2+bm2, sigmoid ----------------
__global__ void head_kernel(const float* __restrict__ pooled, const float* __restrict__ counts,
                            const float* __restrict__ b2,
                            const float* __restrict__ Wm1, const float* __restrict__ bm1,
                            const float* __restrict__ Wm2, const float* __restrict__ bm2,
                            float* __restrict__ out) {
    int g = blockIdx.x * blockDim.x + threadIdx.x;
    if (g >= N_GRAPHS) return;
    float inv = 1.0f / fmaxf(counts[g], 1.0f);
    float r[HIDDEN];
#pragma unroll
    for (int f = 0; f < HIDDEN; ++f)
        r[f] = pooled[(g << 5) + f] * inv + b2[f];   // b2 folded here (linear in mean)
    float acc = bm2[0];
#pragma unroll 4
    for (int j = 0; j < HIDDEN; ++j) {
        float t = bm1[j];
#pragma unroll
        for (int f = 0; f < HIDDEN; ++f)
            t = fmaf(r[f], Wm1[f * HIDDEN + j], t);
        t = fmaxf(t, 0.0f);
        acc = fmaf(t, Wm2[j], acc);
    }
    out[g] = 1.0f / (1.0f + expf(-acc));
}

// ---------------- launch ----------------
extern "C" void kernel_launch(void* const* d_in, const int* in_sizes, int n_in,
                              void* d_out, int out_size, void* d_ws, size_t ws_size,
                              hipStream_t stream) {
    (void)in_sizes; (void)n_in; (void)out_size; (void)ws_size;

    const float* x     = (const float*)d_in[0];
    const int*   ei    = (const int*)d_in[1];     // [2, E]
    const int*   batch = (const int*)d_in[2];
    const float* W1    = (const float*)d_in[3];
    const float* b1    = (const float*)d_in[4];
    const float* W2    = (const float*)d_in[5];
    const float* b2    = (const float*)d_in[6];
    const float* Wm1   = (const float*)d_in[7];
    const float* bm1   = (const float*)d_in[8];
    const float* Wm2   = (const float*)d_in[9];
    const float* bm2   = (const float*)d_in[10];
    float*       out   = (float*)d_out;

    const int* srcI = ei;
    const int* dstI = ei + N_EDGES;

    // workspace layout (floats)
    float* ws     = (float*)d_ws;
    float* dinv   = ws;                                 // N          (deg -> dinv in place)
    float* buf0   = ws + 100352;                        // N*32       (h1, later agg2)
    float* buf1   = buf0 + (size_t)N_NODES * 32;        // N*32       (agg1; bias+relu fused into gemm2)
    float* buf2   = buf1 + (size_t)N_NODES * 32;        // N*32       (h2)
    float* pooled = buf2 + (size_t)N_NODES * 32;        // G*32
    float* counts = pooled + N_GRAPHS * 32;             // G

    const int TPB = 256;
    const int gN   = (N_NODES + TPB - 1) / TPB;
    const int gE   = (N_EDGES + TPB - 1) / TPB;
    const int gNF  = (N_NODES * 32 + TPB - 1) / TPB;
    const long long scat_total = ((long long)N_EDGES + N_NODES) << 5;
    const int gScat = (int)((scat_total + TPB - 1) / TPB);
    const int gemm_blocks = (6250 + 3) / 4;             // 4 wave-tiles per 128-thread block

    // 1) degree (init 1.0 = self-loop), accumulate, rsqrt -> dinv
    fill_kernel<<<gN, TPB, 0, stream>>>(dinv, 1.0f, N_NODES);
    deg_accum_kernel<<<gE, TPB, 0, stream>>>(dstI, dinv, N_EDGES);
    rsqrt_kernel<<<gN, TPB, 0, stream>>>(dinv, N_NODES);

    // 2) h1 = x @ W1   (fp32 WMMA, K=128)
    gemm_n32_wmma_f32<F_IN, false><<<gemm_blocks, 128, 0, stream>>>(x, W1, buf0, N_NODES, nullptr);

    // 3) conv1 aggregate (bias+relu deferred into gemm2's A prologue)
    fill_kernel<<<gNF, TPB, 0, stream>>>(buf1, 0.0f, N_NODES * 32);
    gcn_scatter_kernel<<<gScat, TPB, 0, stream>>>(srcI, dstI, dinv, buf0, buf1, N_EDGES, N_NODES);

    // 4) h2 = relu(agg1 + b1) @ W2   (fp32 WMMA, K=32, fused prologue)
    gemm_n32_wmma_f32<HIDDEN, true><<<gemm_blocks, 128, 0, stream>>>(buf1, W2, buf2, N_NODES, b1);

    // 5) conv2 aggregate (b2 folded into head)
    fill_kernel<<<gNF, TPB, 0, stream>>>(buf0, 0.0f, N_NODES * 32);
    gcn_scatter_kernel<<<gScat, TPB, 0, stream>>>(srcI, dstI, dinv, buf2, buf0, N_EDGES, N_NODES);

    // 6) mean pool
    fill_kernel<<<(N_GRAPHS * 33 + TPB - 1) / TPB, TPB, 0, stream>>>(pooled, 0.0f, N_GRAPHS * 33);
    counts_kernel<<<gN, TPB, 0, stream>>>(batch, counts, N_NODES);
    pool_sum_kernel<<<gNF, TPB, 0, stream>>>(batch, buf0, pooled, N_NODES);

    // 7) MLP head + sigmoid
    head_kernel<<<1, N_GRAPHS, 0, stream>>>(pooled, counts, b2, Wm1, bm1, Wm2, bm2, out);
}